// MemoryModule_19353122636395
// MI455X (gfx1250) — compile-verified
//
#include <hip/hip_runtime.h>
#include <hip/hip_bf16.h>
#include <math.h>

// Problem constants (from reference): N=16384, D=128, P=256
#define NN 16384
#define DD 128
#define PP 256

typedef __attribute__((ext_vector_type(2))) float v2f;
typedef __attribute__((ext_vector_type(8))) float v8f;

// D = A(16x4 f32) * B(4x16 f32) + C(16x16 f32), wave32
// args: (neg_a, A, neg_b, B, c_mod, C, reuse_a, reuse_b)
#define WMMA_F32(a, b, c) \
    __builtin_amdgcn_wmma_f32_16x16x4_f32(false, (a), false, (b), (short)0, (c), false, false)

// ---------------------------------------------------------------------------
// Kernel 1: per 16-row tile: logits GEMM (WMMA), softmax, top-2, distances,
//           output GEMM (WMMA), write [output | repres].
// Block = 128 threads = 4 waves; each wave owns one 16-row tile.
// Grid = N/64 blocks.
// ---------------------------------------------------------------------------
__global__ __launch_bounds__(128) void attn_kernel(
    const float* __restrict__ repres,   // [N][D]
    const float* __restrict__ memory,   // [P][D]
    float* __restrict__ out,            // [N][2D]
    int*   __restrict__ top1w,          // [N]
    float* __restrict__ dist1w,         // [N]
    float* __restrict__ diffw)          // [N]
{
    __shared__ float attn[4][16 * PP];  // 4 waves * 16KB = 64KB

    const int lane = threadIdx.x & 31;
    const int wave = threadIdx.x >> 5;
    const int r0   = blockIdx.x * 64 + wave * 16;

    const int m  = lane & 15;          // A-frag / C-frag row group
    const int n  = lane & 15;          // B-frag column
    const int kb = (lane >> 4) << 1;   // 0 or 2: K sub-offset per half-wave

    float* attnW = &attn[wave][0];

    // memory matrix (128KB) is reused by every wave -> prefetch into caches
    __builtin_prefetch(memory, 0, 3);

    // ---------------- GEMM1: logits[16 x 256] = repres_tile @ memory^T -----
    for (int ct = 0; ct < PP / 16; ++ct) {
        const int p0 = ct * 16;
        v8f acc = {};
        for (int k = 0; k < DD; k += 4) {
            v2f a, b;
            // A: 16x4 tile of repres; lane m holds row m, K = k+kb, k+kb+1
            a.x = repres[(size_t)(r0 + m) * DD + k + kb];
            a.y = repres[(size_t)(r0 + m) * DD + k + kb + 1];
            // B[k][p] = memory[p][k]  (memory^T)
            b.x = memory[(size_t)(p0 + n) * DD + k + kb];
            b.y = memory[(size_t)(p0 + n) * DD + k + kb + 1];
            acc = WMMA_F32(a, b, acc);
        }
        // C/D layout: VGPR v -> row v + 8*(lane>>4), col = lane&15
#pragma unroll
        for (int v = 0; v < 8; ++v) {
            const int row = v + ((lane >> 4) << 3);
            attnW[row * PP + p0 + (lane & 15)] = acc[v];
        }
    }
    __syncthreads();

    // ---------------- softmax prep + top-2 + distances (lanes 0..15) -------
    float rmax = 0.0f, rsum = 1.0f;
    if (lane < 16) {
        const int row = r0 + lane;
        float t1v = -INFINITY, t2v = -INFINITY;
        int   t1i = 0, t2i = 0;
        for (int p = 0; p < PP; ++p) {
            const float l = attnW[lane * PP + p];
            if (l > t1v)      { t2v = t1v; t2i = t1i; t1v = l; t1i = p; }
            else if (l > t2v) { t2v = l;  t2i = p; }
        }
        rmax = t1v;
        float s = 0.0f;
        for (int p = 0; p < PP; ++p)
            s += __expf(attnW[lane * PP + p] - rmax);
        rsum = s;

        // distances to top-1 / top-2 prototypes
        const float* rv = repres + (size_t)row * DD;
        const float* m1 = memory + (size_t)t1i * DD;
        const float* m2 = memory + (size_t)t2i * DD;
        float d1 = 0.0f, d2 = 0.0f;
        for (int d = 0; d < DD; ++d) {
            const float x1 = rv[d] - m1[d];
            const float x2 = rv[d] - m2[d];
            d1 += x1 * x1;
            d2 += x2 * x2;
        }
        d1 = sqrtf(d1) / (float)DD;
        d2 = sqrtf(d2) / (float)DD;
        top1w[row]  = t1i;
        dist1w[row] = d1;
        diffw[row]  = d1 - d2 + 0.001f;
    }
    __syncthreads();

    // ---------------- normalize logits -> attention (all lanes) ------------
    for (int idx = lane; idx < 16 * PP; idx += 32) {
        const int rr = idx >> 8;                 // uniform within each iteration
        const float mxv = __shfl(rmax, rr);      // ds_bpermute broadcast
        const float smv = __shfl(rsum, rr);
        attnW[idx] = __expf(attnW[idx] - mxv) / smv;
    }
    __syncthreads();

    // ---------------- GEMM2: output[16 x 128] = attn @ memory --------------
    for (int ct = 0; ct < DD / 16; ++ct) {
        const int c0 = ct * 16;
        v8f acc = {};
        for (int k = 0; k < PP; k += 4) {
            v2f a, b;
            a.x = attnW[m * PP + k + kb];
            a.y = attnW[m * PP + k + kb + 1];
            b.x = memory[(size_t)(k + kb) * DD + c0 + n];
            b.y = memory[(size_t)(k + kb + 1) * DD + c0 + n];
            acc = WMMA_F32(a, b, acc);
        }
#pragma unroll
        for (int v = 0; v < 8; ++v) {
            const int row = v + ((lane >> 4) << 3);
            out[(size_t)(r0 + row) * (2 * DD) + c0 + (lane & 15)] = acc[v];
        }
    }

    // ---------------- copy repres into out[:, 128:256] ----------------------
    for (int idx = lane; idx < 16 * DD; idx += 32) {
        const int rr = idx >> 7;
        const int dd = idx & (DD - 1);
        out[(size_t)(r0 + rr) * (2 * DD) + DD + dd] =
            repres[(size_t)(r0 + rr) * DD + dd];
    }
}

// ---------------------------------------------------------------------------
// Kernel 2: loss = mean(dist1) + (cnt>0 ? sum(diff<0)/cnt : 0) + ||memory||_F
// One block of 256 threads.
// ---------------------------------------------------------------------------
__global__ __launch_bounds__(256) void loss_kernel(
    const float* __restrict__ memory,
    const float* __restrict__ dist1w,
    const float* __restrict__ diffw,
    float* __restrict__ out_loss)
{
    __shared__ float s1[256];
    __shared__ float s2[256];
    __shared__ int   sc[256];
    __shared__ float sq[256];

    const int t = threadIdx.x;
    float sum1 = 0.0f, sumneg = 0.0f;
    int   cnt  = 0;
    for (int i = t; i < NN; i += 256) {
        sum1 += dist1w[i];
        const float df = diffw[i];
        if (df < 0.0f) { sumneg += df; ++cnt; }
    }
    float ssq = 0.0f;
    for (int i = t; i < PP * DD; i += 256) {
        const float v = memory[i];
        ssq += v * v;
    }
    s1[t] = sum1; s2[t] = sumneg; sc[t] = cnt; sq[t] = ssq;
    __syncthreads();
    for (int s = 128; s > 0; s >>= 1) {
        if (t < s) {
            s1[t] += s1[t + s];
            s2[t] += s2[t + s];
            sc[t] += sc[t + s];
            sq[t] += sq[t + s];
        }
        __syncthreads();
    }
    if (t == 0) {
        float loss = s1[0] / (float)NN;
        if (sc[0] > 0) loss += s2[0] / (float)sc[0];
        loss += sqrtf(sq[0]);
        *out_loss = loss;
    }
}

// ---------------------------------------------------------------------------
// Kernel 3: same_proto_mask[i][j] = (top1[i]==top1[j]) ? 1 : 0
// Bandwidth-dominant kernel (~1.07 GB of stores). Each block stages the full
// top1 array (64KB) in LDS once and emits MROWS rows, so global reads drop
// from ~1.07 GB to ~32 MB. The mask region starts at float offset N*2D+1, so
// row starts are 4B-aligned only; j==3 (mod 4) is 16B-aligned -> write a
// 3-element head, 4095 float4 (b128) bodies, and a 1-element tail per row.
// ---------------------------------------------------------------------------
#define MROWS 32
__global__ __launch_bounds__(256) void mask_kernel(
    const int* __restrict__ top1w,
    float* __restrict__ maskout)   // d_out + N*2D + 1
{
    __shared__ int sTop[NN];       // 64KB

    for (int j = threadIdx.x; j < NN; j += 256)
        sTop[j] = top1w[j];
    __syncthreads();

    const int i0 = blockIdx.x * MROWS;
    for (int r = 0; r < MROWS; ++r) {
        const int i  = i0 + r;
        const int ti = sTop[i];
        float* row = maskout + (size_t)i * NN;

        // head: j = 0,1,2 (unaligned prefix)
        if (threadIdx.x < 3)
            row[threadIdx.x] = (sTop[threadIdx.x] == ti) ? 1.0f : 0.0f;

        // body: 4095 aligned float4 stores covering j = 3 .. 16382
        float4* v4 = (float4*)(row + 3);
        for (int q = threadIdx.x; q < (NN - 4) / 4; q += 256) {
            const int j = 3 + 4 * q;
            float4 val;
            val.x = (sTop[j + 0] == ti) ? 1.0f : 0.0f;
            val.y = (sTop[j + 1] == ti) ? 1.0f : 0.0f;
            val.z = (sTop[j + 2] == ti) ? 1.0f : 0.0f;
            val.w = (sTop[j + 3] == ti) ? 1.0f : 0.0f;
            v4[q] = val;
        }

        // tail: j = NN-1
        if (threadIdx.x == 3)
            row[NN - 1] = (sTop[NN - 1] == ti) ? 1.0f : 0.0f;
    }
}

// ---------------------------------------------------------------------------
extern "C" void kernel_launch(void* const* d_in, const int* in_sizes, int n_in,
                              void* d_out, int out_size, void* d_ws, size_t ws_size,
                              hipStream_t stream)
{
    const float* repres = (const float*)d_in[0];   // [N][D] f32
    const float* memory = (const float*)d_in[1];   // [P][D] f32
    float* out = (float*)d_out;

    // workspace layout
    int*   top1w  = (int*)d_ws;
    float* dist1w = (float*)((char*)d_ws + (size_t)NN * sizeof(int));
    float* diffw  = (float*)((char*)d_ws + (size_t)NN * (sizeof(int) + sizeof(float)));

    attn_kernel<<<NN / 64, 128, 0, stream>>>(repres, memory, out, top1w, dist1w, diffw);

    loss_kernel<<<1, 256, 0, stream>>>(memory, dist1w, diffw,
                                       out + (size_t)NN * (2 * DD));

    mask_kernel<<<NN / MROWS, 256, 0, stream>>>(top1w,
                                                out + (size_t)NN * (2 * DD) + 1);
}